// Float64EnergyLoss_12146167513812
// MI455X (gfx1250) — compile-verified
//
#include <hip/hip_runtime.h>
#include <hip/hip_bf16.h>

// ---------------------------------------------------------------------------
// FEM energy loss (Float64EnergyLoss): bandwidth-bound gather + f64 reduction.
//   Kernel A: u_phys = pred * scales (f64 -> f32 out), W partials (f64)
//   Kernel B: async(global->LDS) double-buffered streaming of element data
//             + L2 gathers of pred rows + f64 energy, Q partials
//   Kernel C: deterministic fixed-order final reduction -> Pi_norm (f32)
// d_out layout: [0] = Pi_norm, [1 .. 3M] = u_phys flat.
// d_ws layout:  [0 .. NBLK_B) = Q partials (double), then NBLK_A W partials.
// ---------------------------------------------------------------------------

#define NTHREADS 256   // 8 wave32 waves per block
#define NBLK_A   1024
#define NBLK_B   2048
#define TILE     NTHREADS   // one element per lane per tile

struct alignas(16) LL2 { long long a, b; };

// LDS byte offset of a generic pointer to __shared__ memory: flat LDS
// addresses are {shared_aperture[63:32], offset[31:0]} and HW truncates to
// addr[31:0] (CDNA5 ISA 10.2), so the low 32 bits ARE the LDS offset.
__device__ __forceinline__ unsigned lds_off(const void* p) {
    return (unsigned)(unsigned long long)p;
}

// Async global->LDS (GVS mode: mem = SADDR64 + VADDR32 + inst_offset),
// tracked by ASYNCcnt.  vdst VGPR carries the LDS byte address.
#define ASYNC_B32(ldsoff, goff, base, imm)                                  \
    asm volatile("global_load_async_to_lds_b32 %0, %1, %2 offset:" #imm    \
                 :: "v"(ldsoff), "v"(goff), "s"(base) : "memory")
#define ASYNC_B128(ldsoff, goff, base)                                      \
    asm volatile("global_load_async_to_lds_b128 %0, %1, %2 offset:0"       \
                 :: "v"(ldsoff), "v"(goff), "s"(base) : "memory")

__device__ __forceinline__ double block_reduce_sum(double v, double* sm) {
    const int tid = threadIdx.x;
    sm[tid] = v;
    __syncthreads();
    #pragma unroll
    for (int s = NTHREADS / 2; s > 0; s >>= 1) {   // fixed order: deterministic
        if (tid < s) sm[tid] += sm[tid + s];
        __syncthreads();
    }
    return sm[0];
}

// ---------------- Kernel A: nodes (flat over 3*N_NODES floats) -------------
__global__ __launch_bounds__(NTHREADS)
void fem_nodes_kernel(const float* __restrict__ pred,
                      const float* __restrict__ fext,
                      const float* __restrict__ uxc,
                      const float* __restrict__ uzc,
                      const float* __restrict__ thc,
                      float* __restrict__ u_out,
                      double* __restrict__ wpart,
                      int n3) {
    const double sux = (double)uxc[0];
    const double suz = (double)uzc[0];
    const double sth = (double)thc[0];

    double acc = 0.0;
    const int stride = gridDim.x * blockDim.x;
    for (int j = blockIdx.x * blockDim.x + threadIdx.x; j < n3; j += stride) {
        const int c = j % 3;
        const double sc = (c == 0) ? sux : (c == 1) ? suz : sth;
        const double u = (double)pred[j] * sc;     // f64, as in reference
        u_out[j] = (float)u;
        acc += (double)fext[j] * u;                // W accumulation (f64)
    }

    __shared__ double sm[NTHREADS];
    const double total = block_reduce_sum(acc, sm);
    if (threadIdx.x == 0) wpart[blockIdx.x] = total;
}

// ---------------- Kernel B: elements (async staging + gather + energy) -----
__global__ __launch_bounds__(NTHREADS)
void fem_elems_kernel(const float* __restrict__ pred,
                      const long long* __restrict__ conn,   // int64 pairs
                      const float* __restrict__ Ls,
                      const float* __restrict__ Es,
                      const float* __restrict__ As,
                      const float* __restrict__ Is,
                      const float* __restrict__ dirs,
                      const float* __restrict__ uxc,
                      const float* __restrict__ uzc,
                      const float* __restrict__ thc,
                      double* __restrict__ qpart,
                      int nelem) {
    __shared__ LL2   sh_conn[2][TILE];
    __shared__ float sh_L[2][TILE], sh_E[2][TILE], sh_A[2][TILE], sh_I[2][TILE];
    __shared__ float sh_c[2][TILE], sh_s[2][TILE];
    __shared__ double sm[NTHREADS];

    const int tid = threadIdx.x;
    const double sux = (double)uxc[0];
    const double suz = (double)uzc[0];
    const double sth = (double)thc[0];

    const int ntiles = (nelem + TILE - 1) / TILE;

    // Stage one tile of streaming element data into this buffer (7 async ops
    // per lane).  Lanes past the end clamp (EXEC stays all-ones; redundant
    // loads of the last element are discarded at accumulate time).
    auto issue_tile = [&](int buf, int tile) {
        int e = tile * TILE + tid;
        const unsigned ec  = (unsigned)((e < nelem) ? e : (nelem - 1));
        const unsigned o16 = ec * 16u;   // conn pair (2 x int64)
        const unsigned o4  = ec * 4u;    // scalar props
        const unsigned o12 = ec * 12u;   // direction row (3 floats)
        ASYNC_B128(lds_off(&sh_conn[buf][tid]), o16, conn);
        ASYNC_B32 (lds_off(&sh_L[buf][tid]),    o4,  Ls,   0);
        ASYNC_B32 (lds_off(&sh_E[buf][tid]),    o4,  Es,   0);
        ASYNC_B32 (lds_off(&sh_A[buf][tid]),    o4,  As,   0);
        ASYNC_B32 (lds_off(&sh_I[buf][tid]),    o4,  Is,   0);
        ASYNC_B32 (lds_off(&sh_c[buf][tid]),    o12, dirs, 0);   // dirs[:,0]
        ASYNC_B32 (lds_off(&sh_s[buf][tid]),    o12, dirs, 8);   // dirs[:,2]
    };

    double acc = 0.0;
    int cur = 0;
    int tile = blockIdx.x;
    if (tile < ntiles) issue_tile(cur, tile);

    for (; tile < ntiles; tile += gridDim.x) {
        const int nxt = tile + gridDim.x;
        if (nxt < ntiles) {
            issue_tile(cur ^ 1, nxt);
            // Per-wave FIFO: <=7 outstanding means current tile's 7 are done.
            asm volatile("s_wait_asynccnt 0x7" ::: "memory");
        } else {
            asm volatile("s_wait_asynccnt 0x0" ::: "memory");
        }
        asm volatile("s_wait_xcnt 0x0" ::: "memory");  // addr-VGPR reuse hazard

        // Each lane consumes exactly the LDS slots it staged -> no barrier.
        const LL2  cn = sh_conn[cur][tid];
        const float Lf = sh_L[cur][tid];
        const float ef = sh_E[cur][tid];
        const float af = sh_A[cur][tid];
        const float if_ = sh_I[cur][tid];
        const double c = (double)sh_c[cur][tid];
        const double s = (double)sh_s[cur][tid];

        const float* __restrict__ pA = pred + 3ll * cn.a;
        const float* __restrict__ pB = pred + 3ll * cn.b;
        __builtin_prefetch(pA, 0, 3);   // global_prefetch_b8 (L2-resident rows)
        __builtin_prefetch(pB, 0, 3);

        const double L  = (double)Lf;
        const double EA = (double)(ef * af);    // f32 product then cast (ref)
        const double EI = (double)(ef * if_);   // f32 product then cast (ref)

        const double uA0 = (double)pA[0] * sux;
        const double uA1 = (double)pA[1] * suz;
        const double tA  = -((double)pA[2] * sth);
        const double uB0 = (double)pB[0] * sux;
        const double uB1 = (double)pB[1] * suz;
        const double tB  = -((double)pB[2] * sth);

        const double u_A =  c * uA0 + s * uA1;
        const double w_A = -s * uA0 + c * uA1;
        const double u_B =  c * uB0 + s * uB1;
        const double w_B = -s * uB0 + c * uB1;

        const double du   = u_A - u_B;
        const double dw   = w_A - w_B;
        const double tsum = tA + tB;

        const double invL  = 1.0 / L;
        const double ea_l  = EA * invL;
        const double ei_l  = EI * invL;
        const double ei_l2 = ei_l * invL;
        const double ei_l3 = ei_l2 * invL;

        const double q = ea_l * du * du
                       + 12.0 * ei_l3 * dw * dw
                       + 12.0 * ei_l2 * dw * tsum
                       + 4.0  * ei_l  * (tA * tA + tB * tB + tA * tB);

        if (tile * TILE + tid < nelem) acc += q;
        cur ^= 1;
    }

    const double total = block_reduce_sum(acc, sm);
    if (tid == 0) qpart[blockIdx.x] = total;
}

// ---------------- Kernel C: deterministic finalize -------------------------
__global__ __launch_bounds__(NTHREADS)
void fem_finalize_kernel(const double* __restrict__ qpart, int nq,
                         const double* __restrict__ wpart, int nw,
                         const float* __restrict__ fcc,
                         const float* __restrict__ uxc,
                         float* __restrict__ out) {
    const int tid = threadIdx.x;

    double q = 0.0;
    for (int i = tid; i < nq; i += NTHREADS) q += qpart[i];   // fixed order
    double w = 0.0;
    for (int i = tid; i < nw; i += NTHREADS) w += wpart[i];   // fixed order

    __shared__ double sq[NTHREADS];
    __shared__ double sw[NTHREADS];
    sq[tid] = q; sw[tid] = w;
    __syncthreads();
    #pragma unroll
    for (int s = NTHREADS / 2; s > 0; s >>= 1) {
        if (tid < s) { sq[tid] += sq[tid + s]; sw[tid] += sw[tid + s]; }
        __syncthreads();
    }

    if (tid == 0) {
        const double U  = 0.5 * sq[0];
        const double W  = sw[0];
        double Ec = (double)(fcc[0] * uxc[0]);   // f32 product then cast (ref)
        if (Ec < 1e-30) Ec = 1e-30;              // jnp.maximum (NaN propagates)
        out[0] = (float)((U - W) / Ec);
    }
}

// ---------------------------------------------------------------------------
extern "C" void kernel_launch(void* const* d_in, const int* in_sizes, int n_in,
                              void* d_out, int out_size, void* d_ws, size_t ws_size,
                              hipStream_t stream) {
    const float*     pred = (const float*)d_in[0];
    const float*     fext = (const float*)d_in[1];
    const long long* conn = (const long long*)d_in[2];
    const float*     Ls   = (const float*)d_in[3];
    const float*     Es   = (const float*)d_in[4];
    const float*     As   = (const float*)d_in[5];
    const float*     Is   = (const float*)d_in[6];
    const float*     dirs = (const float*)d_in[7];
    const float*     uxc  = (const float*)d_in[8];
    const float*     uzc  = (const float*)d_in[9];
    const float*     thc  = (const float*)d_in[10];
    const float*     fcc  = (const float*)d_in[11];

    const int n3    = in_sizes[0];  // 3 * N_NODES
    const int nelem = in_sizes[3];  // N_ELEM

    float* out   = (float*)d_out;
    float* u_out = out + 1;         // u_phys flat, after Pi_norm scalar

    double* qpart = (double*)d_ws;          // NBLK_B doubles
    double* wpart = qpart + NBLK_B;         // NBLK_A doubles

    fem_nodes_kernel<<<NBLK_A, NTHREADS, 0, stream>>>(
        pred, fext, uxc, uzc, thc, u_out, wpart, n3);

    fem_elems_kernel<<<NBLK_B, NTHREADS, 0, stream>>>(
        pred, conn, Ls, Es, As, Is, dirs, uxc, uzc, thc, qpart, nelem);

    fem_finalize_kernel<<<1, NTHREADS, 0, stream>>>(
        qpart, NBLK_B, wpart, NBLK_A, fcc, uxc, out);
}